// TSM_6614249636180
// MI455X (gfx1250) — compile-verified
//
#include <hip/hip_runtime.h>

// Temporal Shift Module (TSM) for x: (B_T=128, C=256, H=56, W=56) fp32,
// num_segments=8, fold_div=3  ->  fold = 85.
//   c in [0, fold)        : out[bt] = x[bt+1] (same c), zero at t == T-1
//   c in [fold, 2*fold)   : out[bt] = x[bt-1] (same c), zero at t == 0
//   c in [2*fold, C)      : out[bt] = x[bt]
// Pure streaming permute: ~822 MB traffic, HBM-bound (~35 us @ 23.3 TB/s).
//
// Grid geometry replaces all div/mod address math:
//   blockIdx.y  = plane = bt*C + c   (32768 planes)
//   blockIdx.x  = which 256-chunk slice of the 784-float4 plane
// The shift decision depends only on blockIdx.y -> wave-uniform -> SALU branch,
// zero lane divergence. Data path: coalesced 128-bit non-temporal load/store
// (411 MB stream each way >> 192 MB L2, zero reuse).

typedef __attribute__((ext_vector_type(4))) float v4f;

constexpr int kBT   = 128;
constexpr int kC    = 256;                 // power of two -> mask/shift
constexpr int kT    = 8;                   // power of two -> mask
constexpr int kFold = kC / 3;              // 85
constexpr int kHW4  = (56 * 56) / 4;       // 784 float4 chunks per plane
constexpr int kPlanes = kBT * kC;          // 32768
constexpr int kXBlocks = (kHW4 + 255) / 256;  // 4

__global__ __launch_bounds__(256)
void tsm_shift_kernel(const float* __restrict__ xf, float* __restrict__ of) {
    const v4f* __restrict__ x = reinterpret_cast<const v4f*>(xf);
    v4f* __restrict__ o       = reinterpret_cast<v4f*>(of);

    const int chunk = blockIdx.x * 256 + threadIdx.x;   // float4 index in plane
    if (chunk >= kHW4) return;                          // only last x-block masks

    const int plane = blockIdx.y;                       // bt*C + c  (uniform)
    const int c     = plane & (kC - 1);
    const int bt    = plane >> 8;                       // log2(kC)
    const int t     = bt & (kT - 1);

    int  src_plane = plane;
    bool zero      = false;
    if (c < kFold) {                 // pull from future frame
        zero      = (t == kT - 1);
        src_plane = plane + kC;      // (bt+1, c)
    } else if (c < 2 * kFold) {      // pull from past frame
        zero      = (t == 0);
        src_plane = plane - kC;      // (bt-1, c)
    }                                // else identity

    v4f v;
    if (zero) {
        v = (v4f)(0.0f);
    } else {
        v = __builtin_nontemporal_load(&x[src_plane * kHW4 + chunk]);
    }
    __builtin_nontemporal_store(v, &o[plane * kHW4 + chunk]);
}

extern "C" void kernel_launch(void* const* d_in, const int* in_sizes, int n_in,
                              void* d_out, int out_size, void* d_ws, size_t ws_size,
                              hipStream_t stream) {
    const float* x = reinterpret_cast<const float*>(d_in[0]);
    float* out     = reinterpret_cast<float*>(d_out);
    (void)in_sizes; (void)n_in; (void)out_size; (void)d_ws; (void)ws_size;

    dim3 grid(kXBlocks, kPlanes, 1);   // (4, 32768)
    dim3 block(256, 1, 1);             // 8 wave32s
    tsm_shift_kernel<<<grid, block, 0, stream>>>(x, out);
}